// SelfAttention_88364657148238
// MI455X (gfx1250) — compile-verified
//
#include <hip/hip_runtime.h>
#include <hip/hip_bf16.h>

typedef float v2f __attribute__((ext_vector_type(2)));
typedef float v8f __attribute__((ext_vector_type(8)));

#define C_   256
#define CK_  32
#define N_   4096

__device__ __forceinline__ v8f wmma_f32(v2f a, v2f b, v8f c) {
    return __builtin_amdgcn_wmma_f32_16x16x4_f32(false, a, false, b, (short)0, c,
                                                 false, false);
}
__device__ __forceinline__ v2f ld2(const float* p) { return *(const v2f*)p; }

// f/g are stored PAIR-INTERLEAVED: element (k, n) -> [(k>>1)*2*N_ + 2*n + (k&1)]
// so a WMMA K-slice (rows k0+2*half, k0+2*half+1 at one column) is ONE aligned
// v2f load, coalesced across lanes (8-byte stride).

// ---------------------------------------------------------------------------
// Kernel 1: fused projections  f = fw@x+fb, g = gw@x+gb, hv = hw@x+hb
// grid: (N/128, 20, B)  block: 256 (8 waves, each a 16x16 output tile, K=256)
// ---------------------------------------------------------------------------
__global__ __launch_bounds__(256) void proj_kernel(
    const float* __restrict__ x,
    const float* __restrict__ fw, const float* __restrict__ fb,
    const float* __restrict__ gw, const float* __restrict__ gb,
    const float* __restrict__ hw, const float* __restrict__ hb,
    float* __restrict__ fbuf, float* __restrict__ gbuf, float* __restrict__ hvbuf)
{
    __shared__ float wtile[16 * 257];   // W tile, pitch 257 -> conflict-free

    const int b     = blockIdx.z;
    const int otile = blockIdx.y;               // 0..19
    const int wv    = threadIdx.x >> 5;
    const int lane  = threadIdx.x & 31;
    const int half  = lane >> 4;
    const int idx   = lane & 15;
    const int n0    = blockIdx.x * 128 + wv * 16;

    const float* W;  const float* bias;  float* outp;  int orow;  bool pairLayout;
    if (otile < 2)      { W = fw; bias = fb; outp = fbuf;  orow = otile * 16;       pairLayout = true;  }
    else if (otile < 4) { W = gw; bias = gb; outp = gbuf;  orow = (otile - 2) * 16; pairLayout = true;  }
    else                { W = hw; bias = hb; outp = hvbuf; orow = (otile - 4) * 16; pairLayout = false; }

    // cooperative, coalesced stage of 16x256 weight tile
    for (int t = threadIdx.x; t < 16 * 256; t += 256) {
        int r = t >> 8, c = t & 255;
        wtile[r * 257 + c] = W[(orow + r) * 256 + c];
    }
    __syncthreads();

    const float* xb = x + (size_t)b * C_ * N_;
    v8f acc = {};
#pragma unroll 4
    for (int k = 0; k < 256; k += 4) {
        v2f a, bb;
        a.x  = wtile[idx * 257 + k + 2 * half];
        a.y  = wtile[idx * 257 + k + 2 * half + 1];
        bb.x = xb[(size_t)(k + 2 * half)     * N_ + n0 + idx];
        bb.y = xb[(size_t)(k + 2 * half + 1) * N_ + n0 + idx];
        acc = wmma_f32(a, bb, acc);
    }

    float* ob = outp + (size_t)b * (pairLayout ? CK_ : C_) * N_;
    if (pairLayout) {
#pragma unroll
        for (int r = 0; r < 8; ++r) {
            int o = orow + r + 8 * half;
            ob[(size_t)(o >> 1) * 2 * N_ + 2 * (n0 + idx) + (o & 1)] = acc[r] + bias[o];
        }
    } else {
#pragma unroll
        for (int r = 0; r < 8; ++r) {
            int o = orow + r + 8 * half;
            ob[(size_t)o * N_ + n0 + idx] = acc[r] + bias[o];
        }
    }
}

// ---------------------------------------------------------------------------
// Kernel 2: row stats of S = f^T g :  M_j = max_m S[j,m], Linv_j = 1/sum exp
// grid: (N/64, B)  block: 128 (4 waves, each owns a 16-row j tile)
// ---------------------------------------------------------------------------
__global__ __launch_bounds__(128) void stats_kernel(
    const float* __restrict__ fbuf, const float* __restrict__ gbuf,
    float* __restrict__ Mrow, float* __restrict__ Linv)
{
    const int b    = blockIdx.y;
    const int wv   = threadIdx.x >> 5;
    const int lane = threadIdx.x & 31;
    const int half = lane >> 4;
    const int idx  = lane & 15;
    const int j0   = blockIdx.x * 64 + wv * 16;

    const float* fp = fbuf + (size_t)b * CK_ * N_;   // pair-interleaved
    const float* gp = gbuf + (size_t)b * CK_ * N_;   // pair-interleaved

    // A-operand (f^T tile) is constant over the m sweep: hoist (b64 coalesced)
    v2f af[8];
#pragma unroll
    for (int k8 = 0; k8 < 8; ++k8)
        af[k8] = ld2(fp + (size_t)(2 * k8 + half) * 2 * N_ + 2 * (j0 + idx));

    float Mv[8], Lv[8];
#pragma unroll
    for (int r = 0; r < 8; ++r) { Mv[r] = -1e30f; Lv[r] = 0.0f; }

    for (int m0 = 0; m0 < N_; m0 += 16) {
        v8f s = {};
#pragma unroll
        for (int k8 = 0; k8 < 8; ++k8) {
            v2f bg = ld2(gp + (size_t)(2 * k8 + half) * 2 * N_ + 2 * (m0 + idx));
            s = wmma_f32(af[k8], bg, s);
        }
#pragma unroll
        for (int r = 0; r < 8; ++r) {
            float v  = s[r];
            float nm = fmaxf(Mv[r], v);
            Lv[r] = Lv[r] * __expf(Mv[r] - nm) + __expf(v - nm);
            Mv[r] = nm;
        }
    }

    // all-reduce (max, sum) over the 16 lanes of each half (rows r+8*half)
#pragma unroll
    for (int r = 0; r < 8; ++r) {
#pragma unroll
        for (int off = 8; off >= 1; off >>= 1) {
            float Mo = __shfl_xor(Mv[r], off, 32);
            float Lo = __shfl_xor(Lv[r], off, 32);
            float nm = fmaxf(Mv[r], Mo);
            Lv[r] = Lv[r] * __expf(Mv[r] - nm) + Lo * __expf(Mo - nm);
            Mv[r] = nm;
        }
    }
    if (idx == 0) {
#pragma unroll
        for (int r = 0; r < 8; ++r) {
            int j = j0 + r + 8 * half;
            Mrow[b * N_ + j] = Mv[r];
            Linv[b * N_ + j] = 1.0f / Lv[r];
        }
    }
}

// ---------------------------------------------------------------------------
// Kernel 3: fused  out = gamma * (hv @ P) + x,  P[j,m] = exp(S[j,m]-M_j)/L_j
// grid: (N/64, B)  block: 256 (8 waves).  WG output = 256c x 64m, j-block 64.
// ---------------------------------------------------------------------------
__global__ __launch_bounds__(256) void attn_kernel(
    const float* __restrict__ x,
    const float* __restrict__ fbuf, const float* __restrict__ gbuf,
    const float* __restrict__ hvbuf,
    const float* __restrict__ Mrow, const float* __restrict__ Linv,
    const float* __restrict__ gammap,
    float* __restrict__ out)
{
    __shared__ float Plds[64 * 64];     // 16 KB P block (rows j, cols m)

    const int b    = blockIdx.y;
    const int m0   = blockIdx.x * 64;
    const int wv   = threadIdx.x >> 5;
    const int lane = threadIdx.x & 31;
    const int half = lane >> 4;
    const int idx  = lane & 15;

    const float* fp = fbuf  + (size_t)b * CK_ * N_;  // pair-interleaved
    const float* gp = gbuf  + (size_t)b * CK_ * N_;  // pair-interleaved
    const float* hp = hvbuf + (size_t)b * C_  * N_;  // row-major [c][n]
    const float* Mr = Mrow + b * N_;
    const float* Lr = Linv + b * N_;

    // This wave's two P tiles: common j sub-tile jt, m sub-tiles mtA, mtA+1
    const int jt  = wv >> 1;            // 0..3
    const int mtA = (wv & 1) * 2;       // 0 or 2

    // g tiles depend only on m0: hoist across the whole j loop (b64 loads)
    v2f gt[2][8];
#pragma unroll
    for (int t = 0; t < 2; ++t)
#pragma unroll
        for (int k8 = 0; k8 < 8; ++k8)
            gt[t][k8] = ld2(gp + (size_t)(2 * k8 + half) * 2 * N_
                               + 2 * (m0 + (mtA + t) * 16 + idx));

    const int cBase = wv * 32;          // this wave's 32 output channels
    v8f acc[2][4];
#pragma unroll
    for (int tc = 0; tc < 2; ++tc)
#pragma unroll
        for (int tm = 0; tm < 4; ++tm) acc[tc][tm] = (v8f){};

    for (int jb = 0; jb < N_; jb += 64) {
        // -------- Phase A: build P[64 x 64] in LDS --------
        const int jrow = jb + jt * 16;
        v2f ft[8];
#pragma unroll
        for (int k8 = 0; k8 < 8; ++k8)
            ft[k8] = ld2(fp + (size_t)(2 * k8 + half) * 2 * N_ + 2 * (jrow + idx));

        float mj[8], lj[8];
#pragma unroll
        for (int r = 0; r < 8; ++r) {
            int j = jrow + r + 8 * half;
            mj[r] = Mr[j];
            lj[r] = Lr[j];
        }
#pragma unroll
        for (int t = 0; t < 2; ++t) {
            v8f s = {};
#pragma unroll
            for (int k8 = 0; k8 < 8; ++k8)
                s = wmma_f32(ft[k8], gt[t][k8], s);
#pragma unroll
            for (int r = 0; r < 8; ++r) {
                float p = __expf(s[r] - mj[r]) * lj[r];
                Plds[(jt * 16 + r + 8 * half) * 64 + (mtA + t) * 16 + idx] = p;
            }
        }
        __syncthreads();

        // -------- Phase B: acc += hv[:, jb:jb+64] x P  (b64 A-loads) --------
#pragma unroll 4
        for (int ks = 0; ks < 64; ks += 4) {
            v2f a0 = ld2(hp + (size_t)(cBase +      idx) * N_ + jb + ks + 2 * half);
            v2f a1 = ld2(hp + (size_t)(cBase + 16 + idx) * N_ + jb + ks + 2 * half);
#pragma unroll
            for (int tm = 0; tm < 4; ++tm) {
                v2f bb;
                bb.x = Plds[(ks + 2 * half)     * 64 + tm * 16 + idx];
                bb.y = Plds[(ks + 2 * half + 1) * 64 + tm * 16 + idx];
                acc[0][tm] = wmma_f32(a0, bb, acc[0][tm]);
                acc[1][tm] = wmma_f32(a1, bb, acc[1][tm]);
            }
        }
        __syncthreads();
    }

    // -------- Epilogue: out = gamma * sa + x --------
    const float gm = gammap[0];
#pragma unroll
    for (int tc = 0; tc < 2; ++tc)
#pragma unroll
        for (int tm = 0; tm < 4; ++tm)
#pragma unroll
            for (int r = 0; r < 8; ++r) {
                int c = cBase + tc * 16 + r + 8 * half;
                int m = m0 + tm * 16 + idx;
                size_t off = ((size_t)b * C_ + c) * N_ + m;
                out[off] = gm * acc[tc][tm][r] + x[off];
            }
}

// ---------------------------------------------------------------------------
extern "C" void kernel_launch(void* const* d_in, const int* in_sizes, int n_in,
                              void* d_out, int out_size, void* d_ws, size_t ws_size,
                              hipStream_t stream) {
    const float* x     = (const float*)d_in[0];
    const float* fw    = (const float*)d_in[1];
    const float* fb    = (const float*)d_in[2];
    const float* gw    = (const float*)d_in[3];
    const float* gb    = (const float*)d_in[4];
    const float* hw    = (const float*)d_in[5];
    const float* hb    = (const float*)d_in[6];
    const float* gamma = (const float*)d_in[7];
    float* out = (float*)d_out;

    const int Bn = in_sizes[0] / (C_ * N_);   // = 4

    float* ws    = (float*)d_ws;
    float* fbuf  = ws;                              // Bn*CK*N (pair-interleaved)
    float* gbuf  = fbuf  + (size_t)Bn * CK_ * N_;   // Bn*CK*N (pair-interleaved)
    float* hvbuf = gbuf  + (size_t)Bn * CK_ * N_;   // Bn*C*N  (row-major)
    float* Mr    = hvbuf + (size_t)Bn * C_  * N_;   // Bn*N
    float* Li    = Mr    + (size_t)Bn * N_;         // Bn*N

    dim3 g1(N_ / 128, (2 * CK_ + C_) / 16, Bn);     // (32, 20, 4)
    proj_kernel<<<g1, 256, 0, stream>>>(x, fw, fb, gw, gb, hw, hb, fbuf, gbuf, hvbuf);

    dim3 g2(N_ / 64, Bn);                           // (64, 4)
    stats_kernel<<<g2, 128, 0, stream>>>(fbuf, gbuf, Mr, Li);

    dim3 g3(N_ / 64, Bn);                           // (64, 4)
    attn_kernel<<<g3, 256, 0, stream>>>(x, fbuf, gbuf, hvbuf, Mr, Li, gamma, out);
}